// SelfAttentionBlock_42726334660749
// MI455X (gfx1250) — compile-verified
//
#include <hip/hip_runtime.h>
#include <stdint.h>

typedef float v2f __attribute__((ext_vector_type(2)));
typedef float v8f __attribute__((ext_vector_type(8)));

constexpr int C_DIM  = 512;
constexpr int HW     = 1024;   // 32*32
constexpr int NB     = 16;
constexpr int GROUPS = 32;
constexpr int CPG    = C_DIM / GROUPS;  // 16
constexpr float EPS  = 1e-6f;

// --- async global->LDS helper (GLOBAL_LOAD_ASYNC_TO_LDS_B128, ASYNCcnt) ---
// Builtin signature (from clang diagnostic): (as1 v4i* src, as3 v4i* dst, Ii, Ii)
typedef int v4i __attribute__((vector_size(16)));
typedef __attribute__((address_space(1))) v4i v4i_g;
typedef __attribute__((address_space(3))) v4i v4i_l;

__device__ __forceinline__ void async_cp16(const void* g, void* l) {
  // generic global addr == AS1 addr; LDS generic addr low 32 bits == LDS offset
  __builtin_amdgcn_global_load_async_to_lds_b128(
      (v4i_g*)(uintptr_t)g,
      (v4i_l*)(uintptr_t)(uint32_t)(uintptr_t)l,
      0, 0);
}

// ---------------------------------------------------------------------------
// Kernel 1: GroupNorm statistics. One block per (n, group); the group slice
// x[n, g*16:(g+1)*16, :, :] is 16384 contiguous floats.
// ---------------------------------------------------------------------------
__global__ void gn_stats_kernel(const float* __restrict__ x,
                                float2* __restrict__ stats) {
  __shared__ float s_sum[256];
  __shared__ float s_sq[256];
  const int ng = blockIdx.x;                       // n*GROUPS + g
  const float4* xv = (const float4*)(x + (size_t)ng * (CPG * HW));
  float sum = 0.f, sq = 0.f;
#pragma unroll
  for (int i = 0; i < 16; ++i) {                   // 16 float4 per thread
    float4 v = xv[threadIdx.x + 256 * i];
    sum += v.x + v.y + v.z + v.w;
    sq  += v.x * v.x + v.y * v.y + v.z * v.z + v.w * v.w;
  }
  s_sum[threadIdx.x] = sum;
  s_sq[threadIdx.x]  = sq;
  __syncthreads();
  for (int off = 128; off > 0; off >>= 1) {
    if (threadIdx.x < off) {
      s_sum[threadIdx.x] += s_sum[threadIdx.x + off];
      s_sq[threadIdx.x]  += s_sq[threadIdx.x + off];
    }
    __syncthreads();
  }
  if (threadIdx.x == 0) {
    const float inv = 1.0f / (float)(CPG * HW);
    float mu  = s_sum[0] * inv;
    float var = s_sq[0] * inv - mu * mu;
    stats[ng] = make_float2(mu, rsqrtf(var + EPS));
  }
}

// ---------------------------------------------------------------------------
// Kernel 2: fused bias Bf[o] = bo[o] + dot(wo[o,:], bv)
// ---------------------------------------------------------------------------
__global__ void fuse_bias_kernel(const float* __restrict__ wo,
                                 const float* __restrict__ bv,
                                 const float* __restrict__ bo,
                                 float* __restrict__ Bf) {
  __shared__ float s_bv[C_DIM];
  const int t = threadIdx.x;
  for (int i = t; i < C_DIM; i += 256) s_bv[i] = bv[i];
  __syncthreads();
  const int o = blockIdx.x * 256 + t;
  const float* w = wo + (size_t)o * C_DIM;
  float acc = 0.f;
  for (int c = 0; c < C_DIM; c += 4) {
    acc += w[c]     * s_bv[c]     + w[c + 1] * s_bv[c + 1] +
           w[c + 2] * s_bv[c + 2] + w[c + 3] * s_bv[c + 3];
  }
  Bf[o] = acc + bo[o];
}

// ---------------------------------------------------------------------------
// Kernel 3: fused weight Wf = wo @ wv  (512x512, fp32 WMMA 16x16x4)
// ---------------------------------------------------------------------------
__global__ void fuse_weight_kernel(const float* __restrict__ wo,
                                   const float* __restrict__ wv,
                                   float* __restrict__ Wf) {
  const int lane  = threadIdx.x & 31;
  const int lmod  = lane & 15;
  const int lhalf = lane >> 4;
  const int tm = blockIdx.y, tn = blockIdx.x;
  const int m  = tm * 16 + lmod;      // output row (A row)
  const int nn = tn * 16 + lmod;      // output col (B col)
  v8f acc = {};
  for (int k0 = 0; k0 < C_DIM; k0 += 4) {
    const int ka = k0 + 2 * lhalf;
    v2f A, B;
    A.x = wo[m * C_DIM + ka];
    A.y = wo[m * C_DIM + ka + 1];
    B.x = wv[ka * C_DIM + nn];
    B.y = wv[(ka + 1) * C_DIM + nn];
    acc = __builtin_amdgcn_wmma_f32_16x16x4_f32(false, A, false, B,
                                                (short)0, acc, false, false);
  }
#pragma unroll
  for (int r = 0; r < 8; ++r)
    Wf[(size_t)(tm * 16 + r + 8 * lhalf) * C_DIM + tn * 16 + lmod] = acc[r];
}

// ---------------------------------------------------------------------------
// Kernel 4: out[n] = Wf @ (a*x[n] + b) + Bf
// Double-buffered GLOBAL_LOAD_ASYNC_TO_LDS_B128 staging; GN affine applied at
// fragment-read time via a per-chunk LDS table.
// ---------------------------------------------------------------------------
constexpr int BM = 128, BN = 128, BK = 16;
constexpr int AS_STRIDE = 20;    // 16B-aligned rows, conflict-free frag reads
constexpr int BS_STRIDE = 136;   // halves land on disjoint bank ranges

__global__ __launch_bounds__(256)
void attn_fused_gemm_kernel(const float* __restrict__ x,
                            const float* __restrict__ gamma,
                            const float* __restrict__ beta,
                            const float2* __restrict__ stats,
                            const float* __restrict__ Wf,
                            const float* __restrict__ Bf,
                            float* __restrict__ out) {
  __shared__ __align__(16) float As[2][BM][AS_STRIDE];   // raw Wf tile [m][k]
  __shared__ __align__(16) float Bs[2][BK][BS_STRIDE];   // raw x  tile [k][p]
  __shared__ float aff_a[2][BK];                         // rstd*gamma per k-row
  __shared__ float aff_b[2][BK];                         // beta - mu*rstd*gamma

  const int lane  = threadIdx.x & 31;
  const int wid   = threadIdx.x >> 5;
  const int lmod  = lane & 15;
  const int lhalf = lane >> 4;
  const int mw = (wid & 3) * 32;        // wave row offset inside BM
  const int nw = (wid >> 2) * 64;       // wave col offset inside BN
  const int n  = blockIdx.z;
  const int mo = blockIdx.y * BM;       // output-channel base
  const int po = blockIdx.x * BN;       // pixel base

  // staging assignments (4x b128 async copies per thread per chunk)
  const int am = threadIdx.x >> 1;             // 0..127  (A row)
  const int ak = (threadIdx.x & 1) * 8;        // 0 or 8  (A k)
  const int bk = threadIdx.x >> 4;             // 0..15   (B k row)
  const int bp = (threadIdx.x & 15) * 8;       // 0..120  (B col)

  const float* gA = Wf + (size_t)(mo + am) * C_DIM + ak;          // + kc
  const float* gX = x + ((size_t)n * C_DIM + bk) * HW + po + bp;  // + kc*HW

  auto stage = [&](int kc, int buf) {
    async_cp16(gA + kc,     &As[buf][am][ak]);
    async_cp16(gA + kc + 4, &As[buf][am][ak + 4]);
    async_cp16(gX + (size_t)kc * HW,     &Bs[buf][bk][bp]);
    async_cp16(gX + (size_t)kc * HW + 4, &Bs[buf][bk][bp + 4]);
    if (threadIdx.x < BK) {
      const int c = kc + threadIdx.x;
      const float2 st = stats[n * GROUPS + (c >> 4)];   // CPG == 16
      const float a = st.y * gamma[c];
      aff_a[buf][threadIdx.x] = a;
      aff_b[buf][threadIdx.x] = beta[c] - st.x * a;
    }
  };

  v8f acc[2][4] = {};

  stage(0, 0);
  for (int kc = 0; kc < C_DIM; kc += BK) {
    const int cur = (kc >> 4) & 1;
    __syncthreads();                    // readers of buf[cur^1] are done
    if (kc + BK < C_DIM) {
      stage(kc + BK, cur ^ 1);
      __builtin_amdgcn_s_wait_asynccnt(4);   // chunk kc landed; next in flight
    } else {
      __builtin_amdgcn_s_wait_asynccnt(0);
    }
    __syncthreads();                    // buf[cur] visible to all waves

#pragma unroll
    for (int ks = 0; ks < BK; ks += 4) {
      const int kk = ks + 2 * lhalf;
      const float a0 = aff_a[cur][kk],     b0 = aff_b[cur][kk];
      const float a1 = aff_a[cur][kk + 1], b1 = aff_b[cur][kk + 1];
      v2f af[2], bf[4];
#pragma unroll
      for (int mt = 0; mt < 2; ++mt) {
        const int row = mw + mt * 16 + lmod;
        af[mt].x = As[cur][row][kk];
        af[mt].y = As[cur][row][kk + 1];
      }
#pragma unroll
      for (int nt = 0; nt < 4; ++nt) {
        const int col = nw + nt * 16 + lmod;
        const float r0 = Bs[cur][kk][col];
        const float r1 = Bs[cur][kk + 1][col];
        bf[nt].x = a0 * r0 + b0;        // GroupNorm affine fused here
        bf[nt].y = a1 * r1 + b1;
      }
#pragma unroll
      for (int mt = 0; mt < 2; ++mt)
#pragma unroll
        for (int nt = 0; nt < 4; ++nt)
          acc[mt][nt] = __builtin_amdgcn_wmma_f32_16x16x4_f32(
              false, af[mt], false, bf[nt], (short)0, acc[mt][nt], false, false);
    }
  }

  // --- epilogue: add fused bias, store ---
#pragma unroll
  for (int mt = 0; mt < 2; ++mt) {
#pragma unroll
    for (int r = 0; r < 8; ++r) {
      const int orow = mo + mw + mt * 16 + r + 8 * lhalf;
      const float bias = Bf[orow];
#pragma unroll
      for (int nt = 0; nt < 4; ++nt) {
        const int p = po + nw + nt * 16 + lmod;
        out[((size_t)n * C_DIM + orow) * HW + p] = acc[mt][nt][r] + bias;
      }
    }
  }
}

// ---------------------------------------------------------------------------
extern "C" void kernel_launch(void* const* d_in, const int* in_sizes, int n_in,
                              void* d_out, int out_size, void* d_ws, size_t ws_size,
                              hipStream_t stream) {
  // inputs: x, gn_gamma, gn_beta, wq, bq, wk, bk, wv, bv, wo, bo
  const float* x     = (const float*)d_in[0];
  const float* gamma = (const float*)d_in[1];
  const float* beta  = (const float*)d_in[2];
  const float* wv    = (const float*)d_in[7];
  const float* bv    = (const float*)d_in[8];
  const float* wo    = (const float*)d_in[9];
  const float* bo    = (const float*)d_in[10];
  float* out = (float*)d_out;

  char* ws = (char*)d_ws;
  float2* stats = (float2*)ws;                  // 512 * 8 B
  float*  Bf    = (float*)(ws + 4096);          // 512 * 4 B
  float*  Wf    = (float*)(ws + 8192);          // 512*512*4 B = 1 MB

  gn_stats_kernel<<<NB * GROUPS, 256, 0, stream>>>(x, stats);
  fuse_bias_kernel<<<C_DIM / 256, 256, 0, stream>>>(wo, bv, bo, Bf);
  fuse_weight_kernel<<<dim3(C_DIM / 16, C_DIM / 16), 32, 0, stream>>>(wo, wv, Wf);
  attn_fused_gemm_kernel<<<dim3(HW / BN, C_DIM / BM, NB), 256, 0, stream>>>(
      x, gamma, beta, stats, Wf, Bf, out);
}